// SortMPNN_51376398795534
// MI455X (gfx1250) — compile-verified
//
#include <hip/hip_runtime.h>
#include <stdint.h>
#include <stddef.h>

#define NN   50000
#define EE   800000
#define DD   128
#define SENT 0xFFFFFFFFu

typedef float v2f __attribute__((ext_vector_type(2)));
typedef float v8f __attribute__((ext_vector_type(8)));

// ---------------------------------------------------------------------------
// K0: blank_p = W @ blank + b   via V_WMMA_F32_16X16X4_F32
// grid = 8 blocks x 32 threads; block rt computes rows [16*rt, 16*rt+16).
// A tile = 16x4 slice of W; B = blank chunk broadcast across all 16 columns
// (every output column is identical; we read column N=0 from lanes 0 and 16).
// ---------------------------------------------------------------------------
__global__ void k_blankp(const float* __restrict__ W, const float* __restrict__ b,
                         const float* __restrict__ blank, float* __restrict__ blank_p) {
  const int lane = threadIdx.x & 31;
  const int rt   = blockIdx.x;       // row tile 0..7
  const int m    = lane & 15;        // A: M index
  const int kh   = lane >> 4;        // K half: 0 -> K{0,1}, 1 -> K{2,3}
  v8f acc = {};
  const float* wrow = W + (rt * 16 + m) * DD;
  for (int kk = 0; kk < DD; kk += 4) {
    v2f A, B;
    A.x = wrow[kk + kh * 2 + 0];
    A.y = wrow[kk + kh * 2 + 1];
    B.x = blank[kk + kh * 2 + 0];
    B.y = blank[kk + kh * 2 + 1];
    acc = __builtin_amdgcn_wmma_f32_16x16x4_f32(false, A, false, B, (short)0, acc,
                                                false, false);
  }
  // D layout: lanes 0-15 -> N=lane, M=vgpr; lanes 16-31 -> N=lane-16, M=vgpr+8
  if (m == 0) {                       // column N==0 lives in lanes 0 and 16
    const int baseM = rt * 16 + kh * 8;
    #pragma unroll
    for (int r = 0; r < 8; ++r) blank_p[baseM + r] = acc[r] + b[baseM + r];
  }
}

// ---------------------------------------------------------------------------
// K1: init per-node top-4 slot arrays to sentinel
// ---------------------------------------------------------------------------
__global__ void k_init(unsigned* __restrict__ slots, int n4) {
  int i = blockIdx.x * blockDim.x + threadIdx.x;
  if (i < n4) slots[i] = SENT;
}

// ---------------------------------------------------------------------------
// K2: concurrent top-4-min insertion of edge ids per destination node.
// Fixed point = 4 smallest edge ids, ascending -> deterministic, and matches
// the reference's "first MAX_NODES edges in stable dst-sort order" semantics
// (the subsequent per-feature sort makes slot order irrelevant).
// ---------------------------------------------------------------------------
__global__ void k_edges(const int* __restrict__ dst, unsigned* __restrict__ slots, int e) {
  int i = blockIdx.x * blockDim.x + threadIdx.x;
  if (i >= e) return;
  unsigned cur = (unsigned)i;
  unsigned* s = slots + (size_t)dst[i] * 4;
  #pragma unroll
  for (int k = 0; k < 4; ++k) {
    unsigned old = atomicMin(&s[k], cur);
    if (old == SENT) return;        // landed in an empty slot, nothing displaced
    cur = old > cur ? old : cur;    // carry the displaced (larger) value down
  }
}

// ---------------------------------------------------------------------------
// K3: per-node message combine. One wave (32 lanes) per node; lane covers
// features [4*lane, 4*lane+4). Gather up to 4 source rows (L2-resident),
// per-feature sort-4 network, dot with cw, h = a*x + out + cb.
// Also emits per-block partial sum/sumsq (fixed-order, deterministic).
// ---------------------------------------------------------------------------
__global__ void k_node(const float* __restrict__ x, const int* __restrict__ src,
                       const unsigned* __restrict__ slots, const float* __restrict__ blank_p,
                       const float* __restrict__ cw, const float* __restrict__ cb,
                       const float* __restrict__ a, float* __restrict__ h,
                       float* __restrict__ partials, int n) {
  __shared__ float lh[8][DD];
  const int wave = threadIdx.x >> 5;
  const int lane = threadIdx.x & 31;
  const int v    = blockIdx.x * 8 + wave;
  const int d0   = lane * 4;
  if (v < n) {
    uint4 s = ((const uint4*)slots)[v];
    unsigned ss[4] = {s.x, s.y, s.z, s.w};
    float4 bp = ((const float4*)blank_p)[lane];
    float vals[4][4];
    #pragma unroll
    for (int mth = 0; mth < 4; ++mth) {
      float4 t;
      if (ss[mth] != SENT) {
        int sr = src[ss[mth]];
        t = ((const float4*)(x + (size_t)sr * DD))[lane];
      } else {
        t = bp;
      }
      vals[mth][0] = t.x; vals[mth][1] = t.y; vals[mth][2] = t.z; vals[mth][3] = t.w;
    }
    const float w0 = cw[0], w1 = cw[1], w2 = cw[2], w3 = cw[3];
    const float av = a[0], cbv = cb[0];
    float4 xr = ((const float4*)(x + (size_t)v * DD))[lane];
    const float* xp = &xr.x;
    float4 hq;
    float* hp = &hq.x;
    #pragma unroll
    for (int c = 0; c < 4; ++c) {
      float p0 = vals[0][c], p1 = vals[1][c], p2 = vals[2][c], p3 = vals[3][c];
      // sort-4 network (ascending): (01)(23)(02)(13)(12)
      float l0 = fminf(p0, p1), h0 = fmaxf(p0, p1);
      float l1 = fminf(p2, p3), h1 = fmaxf(p2, p3);
      float m0 = fminf(l0, l1), t0 = fmaxf(l0, l1);
      float t1 = fminf(h0, h1), m3 = fmaxf(h0, h1);
      float m1 = fminf(t0, t1), m2 = fmaxf(t0, t1);
      float out = w0 * m0 + w1 * m1 + w2 * m2 + w3 * m3 + cbv;
      float hv  = av * xp[c] + out;
      hp[c] = hv;
      lh[wave][d0 + c] = hv;
    }
    ((float4*)(h + (size_t)v * DD))[lane] = hq;
  } else {
    #pragma unroll
    for (int c = 0; c < 4; ++c) lh[wave][d0 + c] = 0.0f;
  }
  __syncthreads();
  if (wave == 0) {
    #pragma unroll
    for (int c = 0; c < 4; ++c) {
      int d = d0 + c;
      float sAcc = 0.f, qAcc = 0.f;
      for (int w = 0; w < 8; ++w) { float t = lh[w][d]; sAcc += t; qAcc += t * t; }
      partials[(size_t)blockIdx.x * 256 + d]       = sAcc;
      partials[(size_t)blockIdx.x * 256 + 128 + d] = qAcc;
    }
  }
}

// ---------------------------------------------------------------------------
// K4: final BN reduction, fixed order, double accumulation. 1 block x 128.
// stats[d] = mu, stats[128+d] = 1/sqrt(var + eps)   (biased var, as jnp.var)
// ---------------------------------------------------------------------------
__global__ void k_bnstats(const float* __restrict__ partials, int nBlocks,
                          float* __restrict__ stats, int n) {
  int d = threadIdx.x;  // 0..127
  double s = 0.0, q = 0.0;
  for (int b = 0; b < nBlocks; ++b) {
    s += (double)partials[(size_t)b * 256 + d];
    q += (double)partials[(size_t)b * 256 + 128 + d];
  }
  double mu  = s / (double)n;
  double var = q / (double)n - mu * mu;
  stats[d]       = (float)mu;
  stats[128 + d] = (float)(1.0 / sqrt(var + 1e-5));
}

// ---------------------------------------------------------------------------
// K5: x_out = x_in + (h - mu) * inv * g + be     (float4 over N*D)
// ---------------------------------------------------------------------------
__global__ void k_apply(const float* __restrict__ x_in, const float* __restrict__ h,
                        const float* __restrict__ stats, const float* __restrict__ g,
                        const float* __restrict__ be, float* __restrict__ x_out, int n4) {
  int i = blockIdx.x * blockDim.x + threadIdx.x;
  if (i >= n4) return;
  int d0 = (i & 31) * 4;  // 32 float4 per 128-feature row
  float4 xv = ((const float4*)x_in)[i];
  float4 hv = ((const float4*)h)[i];
  const float* xp = &xv.x;
  const float* hp = &hv.x;
  float4 o;
  float* op = &o.x;
  #pragma unroll
  for (int c = 0; c < 4; ++c) {
    int d = d0 + c;
    op[c] = xp[c] + (hp[c] - stats[d]) * stats[128 + d] * g[d] + be[d];
  }
  ((float4*)x_out)[i] = o;
}

// ---------------------------------------------------------------------------
extern "C" void kernel_launch(void* const* d_in, const int* in_sizes, int n_in,
                              void* d_out, int out_size, void* d_ws, size_t ws_size,
                              hipStream_t stream) {
  (void)in_sizes; (void)n_in; (void)out_size; (void)ws_size;

  const float* x0  = (const float*)d_in[0];
  const int*   src = (const int*)d_in[1];        // edge_index row 0
  const int*   dst = (const int*)d_in[1] + EE;   // edge_index row 1

  // workspace carve-up (all 16B aligned)
  uint8_t* ws = (uint8_t*)d_ws;
  unsigned* slots   = (unsigned*)ws;                         // NN*4 u32 = 800000 B
  float*    blank_p = (float*)(ws + (size_t)NN * 4 * 4);     // 128 f
  float*    stats   = blank_p + 128;                         // 256 f
  float*    h       = stats + 256;                           // NN*DD f
  float*    partials= h + (size_t)NN * DD;                   // nBlocks*256 f

  const int nBlocks = (NN + 7) / 8;          // 6250
  const int n4      = NN * DD / 4;           // 1.6M float4
  float* x_out = (float*)d_out;

  for (int layer = 0; layer < 2; ++layer) {
    const int base = 2 + 8 * layer;
    const float* W     = (const float*)d_in[base + 0];
    const float* b     = (const float*)d_in[base + 1];
    const float* cw    = (const float*)d_in[base + 2];
    const float* cb    = (const float*)d_in[base + 3];
    const float* a     = (const float*)d_in[base + 4];
    const float* g     = (const float*)d_in[base + 5];
    const float* be    = (const float*)d_in[base + 6];
    const float* blank = (const float*)d_in[base + 7];
    const float* xl    = (layer == 0) ? x0 : x_out;

    k_blankp<<<8, 32, 0, stream>>>(W, b, blank, blank_p);
    k_init  <<<(NN * 4 + 255) / 256, 256, 0, stream>>>(slots, NN * 4);
    k_edges <<<(EE + 255) / 256, 256, 0, stream>>>(dst, slots, EE);
    k_node  <<<nBlocks, 256, 0, stream>>>(xl, src, slots, blank_p, cw, cb, a,
                                          h, partials, NN);
    k_bnstats<<<1, 128, 0, stream>>>(partials, nBlocks, stats, NN);
    k_apply <<<(n4 + 255) / 256, 256, 0, stream>>>(xl, h, stats, g, be, x_out, n4);
  }
}